// DifferentiablePersistenceLandscape_58755152609838
// MI455X (gfx1250) — compile-verified
//
#include <hip/hip_runtime.h>

// Problem constants (match the reference)
#define RESOLUTION 256     // t-grid size
#define BATCH      64
#define HDIMS      3
#define PPTS       2048    // points per diagram
#define MAXPERS    2.0f

// Parallel decomposition
#define TPB        256                    // threads per block (8 wave32s)
#define SPLIT      4                      // blocks per diagram (t-grid split)
#define SUBS       4                      // lanes cooperating per t value
#define TVALS      (RESOLUTION / SPLIT)   // 64 t values per block
#define PPT        (PPTS / SUBS)          // 512 points per thread

// ---- CDNA5 async global->LDS path detection -------------------------------
#if defined(__has_builtin)
#if __has_builtin(__builtin_amdgcn_global_load_async_to_lds_b128)
#define HAVE_ASYNC_LDS 1
#endif
#endif

// Builtin signature is (v4i addrspace(1)*, v4i addrspace(3)*, imm, imm).
typedef int v4i __attribute__((ext_vector_type(4)));
typedef v4i __attribute__((address_space(1)))* as1_v4i_ptr;  // global
typedef v4i __attribute__((address_space(3)))* as3_v4i_ptr;  // LDS

// Branchless sorted insert into a descending top-5 register file.
// Keeps the 5 largest values of the streamed multiset (all values >= 0).
__device__ __forceinline__ void ins5(float h, float& a0, float& a1,
                                     float& a2, float& a3, float& a4) {
  float m;
  m = fmaxf(a0, h); h = fminf(a0, h); a0 = m;
  m = fmaxf(a1, h); h = fminf(a1, h); a1 = m;
  m = fmaxf(a2, h); h = fminf(a2, h); a2 = m;
  m = fmaxf(a3, h); h = fminf(a3, h); a3 = m;
  a4 = fmaxf(a4, h);
}

__global__ __launch_bounds__(TPB)
void pl_topk_kernel(const float* __restrict__ births,
                    const float* __restrict__ deaths,
                    const float* __restrict__ lweights,
                    const float* __restrict__ pscale,
                    float* __restrict__ out) {
  // 16 KB of LDS per workgroup: one full diagram, reused 64x by the block.
  __shared__ __align__(16) float sB[PPTS];
  __shared__ __align__(16) float sD[PPTS];

  const int bd  = blockIdx.x / SPLIT;   // 0 .. B*D-1  (diagram)
  const int q   = blockIdx.x % SPLIT;   // 0 .. 3      (t-grid quarter)
  const int tid = threadIdx.x;          // 0 .. 255
  const int tl  = tid >> 2;             // 0 .. 63     (local t index)
  const int sub = tid & 3;              // 0 .. 3      (P-axis subset)

  const float* gB = births + (size_t)bd * PPTS;
  const float* gD = deaths + (size_t)bd * PPTS;

#if defined(HAVE_ASYNC_LDS)
  // Each thread issues 4x async B128 copies (16 B each): 256 threads cover
  // 2 x 8 KB. ASYNCcnt tracks completion; per-wave drain then block barrier.
  __builtin_amdgcn_global_load_async_to_lds_b128(
      (as1_v4i_ptr)(gB + tid * 4),        (as3_v4i_ptr)(sB + tid * 4), 0, 0);
  __builtin_amdgcn_global_load_async_to_lds_b128(
      (as1_v4i_ptr)(gB + 1024 + tid * 4), (as3_v4i_ptr)(sB + 1024 + tid * 4), 0, 0);
  __builtin_amdgcn_global_load_async_to_lds_b128(
      (as1_v4i_ptr)(gD + tid * 4),        (as3_v4i_ptr)(sD + tid * 4), 0, 0);
  __builtin_amdgcn_global_load_async_to_lds_b128(
      (as1_v4i_ptr)(gD + 1024 + tid * 4), (as3_v4i_ptr)(sD + 1024 + tid * 4), 0, 0);
#if __has_builtin(__builtin_amdgcn_s_wait_asynccnt)
  __builtin_amdgcn_s_wait_asynccnt(0);
#else
  asm volatile("s_wait_asynccnt 0" ::: "memory");
#endif
  __syncthreads();
#else
  // Fallback: vectorized synchronous staging.
  ((float4*)sB)[tid]       = ((const float4*)gB)[tid];
  ((float4*)sB)[256 + tid] = ((const float4*)gB)[256 + tid];
  ((float4*)sD)[tid]       = ((const float4*)gD)[tid];
  ((float4*)sD)[256 + tid] = ((const float4*)gD)[256 + tid];
  __syncthreads();
#endif

  // Softmax over the K=5 landscape weights (uniform; compiler scalarizes).
  const float w0 = lweights[0], w1 = lweights[1], w2 = lweights[2],
              w3 = lweights[3], w4 = lweights[4];
  const float mx = fmaxf(fmaxf(fmaxf(w0, w1), fmaxf(w2, w3)), w4);
  const float e0 = __expf(w0 - mx), e1 = __expf(w1 - mx), e2 = __expf(w2 - mx),
              e3 = __expf(w3 - mx), e4 = __expf(w4 - mx);
  const float inv = 1.0f / (e0 + e1 + e2 + e3 + e4);
  const float scale = pscale[0];

  // linspace(0, MAXPERS, RESOLUTION): t_r = r * MAXPERS/(RES-1)
  const int   tr = q * TVALS + tl;
  const float t  = (float)tr * (MAXPERS / (float)(RESOLUTION - 1));

  float a0 = 0.0f, a1 = 0.0f, a2 = 0.0f, a3 = 0.0f, a4 = 0.0f;

  const float4* b4 = (const float4*)sB;
  const float4* d4 = (const float4*)sD;

  // 4 cooperating lanes (sub=0..3) split the 2048 points: lane `sub` reads
  // float4 index 4*i + sub -> the 4 lanes hit 4 consecutive 16B chunks
  // (conflict-free, broadcast to the 8 t-groups of the wave).
#pragma unroll 2
  for (int i = 0; i < PPTS / (4 * SUBS); ++i) {   // 128 iterations, 512 pts
    const int p4 = (i << 2) | sub;
    const float4 bb = b4[p4];
    const float4 dd = d4[p4];
    float h;
    h = fmaxf(fminf(t - bb.x, dd.x - t), 0.0f); ins5(h, a0, a1, a2, a3, a4);
    h = fmaxf(fminf(t - bb.y, dd.y - t), 0.0f); ins5(h, a0, a1, a2, a3, a4);
    h = fmaxf(fminf(t - bb.z, dd.z - t), 0.0f); ins5(h, a0, a1, a2, a3, a4);
    h = fmaxf(fminf(t - bb.w, dd.w - t), 0.0f); ins5(h, a0, a1, a2, a3, a4);
  }

  // Merge the 4 partial top-5 lists across the cooperating lanes.
  // top5(top5(X) U top5(Y)) == top5(X U Y), so two xor rounds are exact.
#pragma unroll
  for (int m = 1; m <= 2; m <<= 1) {
    const float p0 = __shfl_xor(a0, m);
    const float p1 = __shfl_xor(a1, m);
    const float p2 = __shfl_xor(a2, m);
    const float p3 = __shfl_xor(a3, m);
    const float p4 = __shfl_xor(a4, m);
    ins5(p0, a0, a1, a2, a3, a4);
    ins5(p1, a0, a1, a2, a3, a4);
    ins5(p2, a0, a1, a2, a3, a4);
    ins5(p3, a0, a1, a2, a3, a4);
    ins5(p4, a0, a1, a2, a3, a4);
  }

  // a0..a4 hold the 1st..5th largest heights (descending), matching top_k.
  if (sub == 0) {
    const float combined =
        (a0 * e0 + a1 * e1 + a2 * e2 + a3 * e3 + a4 * e4) * inv * scale;
    out[bd * RESOLUTION + tr] = combined;
  }
}

extern "C" void kernel_launch(void* const* d_in, const int* in_sizes, int n_in,
                              void* d_out, int out_size, void* d_ws,
                              size_t ws_size, hipStream_t stream) {
  const float* births  = (const float*)d_in[0];
  const float* deaths  = (const float*)d_in[1];
  const float* lw      = (const float*)d_in[2];
  const float* pscale  = (const float*)d_in[3];
  float*       out     = (float*)d_out;

  const int nblocks = BATCH * HDIMS * SPLIT;  // 768 workgroups
  pl_topk_kernel<<<nblocks, TPB, 0, stream>>>(births, deaths, lw, pscale, out);
}